// ActQuantizer_57105885168130
// MI455X (gfx1250) — compile-verified
//
#include <hip/hip_runtime.h>
#include <stdint.h>

#ifndef __has_builtin
#define __has_builtin(x) 0
#endif

#define FP4_MAX   6.0f
#define FP8_MAX   448.0f
#define NVFP4_MAX (FP4_MAX * FP8_MAX)   /* 2688 */

typedef float f4 __attribute__((ext_vector_type(4)));

// ---------------------------------------------------------------------------
// Helpers
// ---------------------------------------------------------------------------

// Round positive v onto the fp8 e4m3fn grid (RNE), mimicking ml_dtypes:
//   overflow (>= 464 = midpoint past 448) -> NaN, underflow (< 2^-10) -> 0.
// Quantum is a power of two, so v*qinv is an exact scaling and rintf gives RNE.
__device__ __forceinline__ float e4m3_roundtrip_pos(float v) {
  if (!(v < 464.0f)) return __builtin_nanf("");   // covers inf / NaN / overflow
  float q, qinv;
  if (v >= 0x1p-6f) {                              // normal range: quantum 2^(e-3)
    int e = (int)(__float_as_uint(v) >> 23) - 127; // floor(log2 v)
    q    = __uint_as_float((unsigned)(e - 3 + 127) << 23);
    qinv = __uint_as_float((unsigned)(127 - (e - 3)) << 23);
  } else {                                         // subnormal grid: quantum 2^-9
    q = 0x1p-9f; qinv = 0x1p9f;
  }
  return rintf(v * qinv) * q;                      // v_rndne_f32 (RNE)
}

// Positive fp4 (1-2-1) grid: spacing 0.5 below 2, 1.0 in [2,4), 2.0 above.
__device__ __forceinline__ float fp4_121_pos(float y) {
  float r1 = rintf(2.0f * y) * 0.5f;
  float r2 = rintf(y);
  float r3 = 2.0f * rintf(0.5f * y);
  return (y < 2.0f) ? r1 : ((y < 4.0f) ? r2 : r3);
}

// ---------------------------------------------------------------------------
// Pass 0: zero the amax accumulator (workspace is poisoned with 0xAA)
// ---------------------------------------------------------------------------
__global__ void zero_amax_kernel(unsigned* __restrict__ amax_bits) {
  *amax_bits = 0u;
}

// ---------------------------------------------------------------------------
// Pass 1: global amax via b128 streaming + wave32 reduce + u32 atomic max
// (non-negative float bit pattern compares like unsigned -> deterministic).
// Regular-temporal loads on purpose: they install the tail of the tensor
// into the 192MB L2 for pass 2 to consume.
// ---------------------------------------------------------------------------
__global__ __launch_bounds__(256) void amax_kernel(const float4* __restrict__ x,
                                                   unsigned* __restrict__ amax_bits,
                                                   int n4) {
  const int tid    = blockIdx.x * blockDim.x + threadIdx.x;
  const int stride = gridDim.x * blockDim.x;
  float m = 0.0f;
  for (int i = tid; i < n4; i += stride) {
    const int pf = i + 4 * stride;                 // 64MB look-ahead, in-bounds only
    if (pf < n4) __builtin_prefetch(&x[pf], 0, 1); // global_prefetch_b8
    float4 v = x[i];
    m = fmaxf(m, fmaxf(fmaxf(fabsf(v.x), fabsf(v.y)),
                       fmaxf(fabsf(v.z), fabsf(v.w))));
  }
#pragma unroll
  for (int off = 16; off > 0; off >>= 1)
    m = fmaxf(m, __shfl_xor(m, off, 32));

  __shared__ float smax[8];                        // 256 threads = 8 wave32
  const int lane = threadIdx.x & 31;
  const int wave = threadIdx.x >> 5;
  if (lane == 0) smax[wave] = m;
  __syncthreads();
  if (wave == 0) {
    m = (lane < ((int)blockDim.x >> 5)) ? smax[lane] : 0.0f;
#pragma unroll
    for (int off = 4; off > 0; off >>= 1)
      m = fmaxf(m, __shfl_xor(m, off, 32));
    if (lane == 0) atomicMax(amax_bits, __float_as_uint(m));
  }
}

// ---------------------------------------------------------------------------
// Pass 2: quantize, traversing in REVERSE so the early iterations hit the
// ~192MB tail of the tensor that pass 1 left resident in L2. One float4 per
// lane; 4 consecutive lanes = one 16-elem block (two shuffle hops for the
// block max). Reversal is done at 4-float4-group granularity to keep that
// lane grouping intact. Loads/stores are non-temporal (read-once data /
// write-only results) so they don't evict the hot tail.
// ---------------------------------------------------------------------------
__global__ __launch_bounds__(256) void quant_kernel(const float4* __restrict__ x,
                                                    float4* __restrict__ out,
                                                    const unsigned* __restrict__ amax_bits,
                                                    int n4) {
#if __has_builtin(__builtin_amdgcn_global_load_async_to_lds_b32) && \
    __has_builtin(__builtin_amdgcn_s_wait_asynccnt)
  // CDNA5 async global->LDS broadcast of the per-tensor amax.
  __shared__ unsigned s_amax;
  if (threadIdx.x == 0) {
    __builtin_amdgcn_global_load_async_to_lds_b32(
        (__attribute__((address_space(1))) int*)amax_bits,
        (__attribute__((address_space(3))) int*)&s_amax, 0, 0);
    __builtin_amdgcn_s_wait_asynccnt(0);
  }
  __syncthreads();
  const float amax = __uint_as_float(s_amax);
#else
  const float amax = __uint_as_float(*amax_bits);  // uniform -> s_load_b32
#endif

  const float scale_t = amax / NVFP4_MAX;          // per-tensor scale
  const float inv_t   = 1.0f / scale_t;

  const int tid    = blockIdx.x * blockDim.x + threadIdx.x;
  const int stride = gridDim.x * blockDim.x;       // multiple of 4

  for (int i = tid; i < n4; i += stride) {
    // reverse mapping at aligned-group-of-4 granularity
    const int j = (n4 - 4) - (i & ~3) + (i & 3);
    const int pj = j - 2 * stride;                 // descending look-ahead
    if (pj >= 0) __builtin_prefetch(&x[pj], 0, 1);

    f4 v = __builtin_nontemporal_load((const f4*)&x[j]);
    f4 a;
    a.x = fabsf(v.x); a.y = fabsf(v.y); a.z = fabsf(v.z); a.w = fabsf(v.w);

    // block (16-elem) max across the 4-lane group
    float bmax = fmaxf(fmaxf(a.x, a.y), fmaxf(a.z, a.w));
    bmax = fmaxf(bmax, __shfl_xor(bmax, 1, 32));
    bmax = fmaxf(bmax, __shfl_xor(bmax, 2, 32));

    // inverse block scale, round-tripped through fp8 e4m3 (invalid -> 1.0)
    const float bmax_s = bmax * inv_t;
    float sb = e4m3_roundtrip_pos(FP4_MAX / bmax_s);
    sb = (sb > 0.0f && sb <= FP8_MAX) ? sb : 1.0f;

    const float k  = inv_t * sb;                   // into fp4 domain
    const float ko = scale_t / sb;                 // back out

    f4 o;
    o.x = __builtin_copysignf(fp4_121_pos(a.x * k) * ko, v.x);
    o.y = __builtin_copysignf(fp4_121_pos(a.y * k) * ko, v.y);
    o.z = __builtin_copysignf(fp4_121_pos(a.z * k) * ko, v.z);
    o.w = __builtin_copysignf(fp4_121_pos(a.w * k) * ko, v.w);
    __builtin_nontemporal_store(o, (f4*)&out[j]);  // global_store_b128 (NT)
  }
}

// ---------------------------------------------------------------------------
// Launcher
// ---------------------------------------------------------------------------
extern "C" void kernel_launch(void* const* d_in, const int* in_sizes, int n_in,
                              void* d_out, int out_size, void* d_ws, size_t ws_size,
                              hipStream_t stream) {
  (void)n_in; (void)out_size; (void)ws_size;
  const float4* x   = (const float4*)d_in[0];
  float4*       out = (float4*)d_out;
  unsigned* amax_bits = (unsigned*)d_ws;

  const int n  = in_sizes[0];       // 4*4096*4096 = 67,108,864
  const int n4 = n >> 2;            // 16,777,216 float4s

  zero_amax_kernel<<<1, 1, 0, stream>>>(amax_bits);
  // 4096*256 = 1,048,576 threads -> 16 float4 each (exact division)
  amax_kernel<<<4096, 256, 0, stream>>>(x, amax_bits, n4);
  // 8192*256 = 2,097,152 threads -> 8 float4 each (exact division)
  quant_kernel<<<8192, 256, 0, stream>>>(x, out, amax_bits, n4);
}